// PScanTriton_8804682956941
// MI455X (gfx1250) — compile-verified
//
#include <hip/hip_runtime.h>
#include <cstdint>

// Problem constants (match reference: B=8, L=4096, D=256).
constexpr int B = 8;
constexpr int L = 4096;
constexpr int D = 256;
constexpr int T = 32;           // timesteps per chunk
constexpr int C = L / T;        // 128 chunks
constexpr int S = 4;            // timesteps per LDS stage
constexpr int NST = T / S;      // 8 stages per chunk

typedef float v4f __attribute__((ext_vector_type(4)));
typedef float v2f __attribute__((ext_vector_type(2)));
typedef uint32_t u32x4 __attribute__((ext_vector_type(4)));
typedef uint32_t u32x8 __attribute__((ext_vector_type(8)));

// ---------------- Phase 1 staging: ASYNCcnt path -------
// Each lane async-copies exactly the float4 rows it will later read, so
// s_wait_asynccnt alone orders LDS visibility (no barrier needed).
__device__ __forceinline__ void async_stage(const float* __restrict__ g0,
                                            const float* __restrict__ g1,
                                            const float* __restrict__ g2,
                                            const float* __restrict__ g3,
                                            const float* l0, const float* l1,
                                            const float* l2, const float* l3,
                                            int s, int d4) {
  const uint32_t lb0 = (uint32_t)(uintptr_t)l0;
  const uint32_t lb1 = (uint32_t)(uintptr_t)l1;
  const uint32_t lb2 = (uint32_t)(uintptr_t)l2;
  const uint32_t lb3 = (uint32_t)(uintptr_t)l3;
#pragma unroll
  for (int r = 0; r < S; ++r) {
    const uint32_t goff = (uint32_t)(((s * S + r) * D + d4) * sizeof(float));
    const uint32_t loff = (uint32_t)((r * D + d4) * sizeof(float));
    asm volatile("global_load_async_to_lds_b128 %0, %1, %2"
                 :: "v"(lb0 + loff), "v"(goff), "s"(g0) : "memory");
    asm volatile("global_load_async_to_lds_b128 %0, %1, %2"
                 :: "v"(lb1 + loff), "v"(goff), "s"(g1) : "memory");
    asm volatile("global_load_async_to_lds_b128 %0, %1, %2"
                 :: "v"(lb2 + loff), "v"(goff), "s"(g2) : "memory");
    asm volatile("global_load_async_to_lds_b128 %0, %1, %2"
                 :: "v"(lb3 + loff), "v"(goff), "s"(g3) : "memory");
  }
}

// ---------------- Phase 3 staging: TDM (TENSORcnt) path --------------------
// One tensor_load_to_lds per array per wave per stage: a 2D S x 128-element
// f32 tile (row stride D elements) DMA'd into LDS, packed 128 floats/row.
// D# layout per CDNA5 ISA.
__device__ __forceinline__ void tdm_load_tile(const float* gsrc, uint32_t lds_byte) {
  const uint64_t ga = (uint64_t)(uintptr_t)gsrc;
  u32x4 g0;
  g0[0] = 1u;                                             // count=1, no gather
  g0[1] = lds_byte;                                       // lds_addr
  g0[2] = (uint32_t)ga;                                   // global_addr[31:0]
  g0[3] = ((uint32_t)(ga >> 32) & 0x01FFFFFFu)            // global_addr[56:32]
          | 0x80000000u;                                  // type=2 ("image")
  u32x8 g1;
  g1[0] = 0x00020000u;           // workgroup_mask=0, data_size=2 (4 bytes)
  g1[1] = (128u << 16);          // atomic_barrier_addr=0 | tensor_dim0[15:0]=128
  g1[2] = ((uint32_t)S << 16);   // tensor_dim0[31:16]=0  | tensor_dim1[15:0]=S
  g1[3] = (128u << 16);          // tensor_dim1[31:16]=0  | tile_dim0=128
  g1[4] = (uint32_t)S;           // tile_dim1=S, tile_dim2=0
  g1[5] = (uint32_t)D;           // tensor_dim0_stride[31:0]=256
  g1[6] = 0u;                    // stride hi | tensor_dim1_stride lo
  g1[7] = 0u;
  asm volatile("tensor_load_to_lds %0, %1" :: "s"(g0), "s"(g1) : "memory");
}

// ---------------- Phase 1: per-chunk local scan -> carries -----------------
__global__ __launch_bounds__(64) void scan_partial(
    const float* __restrict__ Are, const float* __restrict__ Aim,
    const float* __restrict__ Xre, const float* __restrict__ Xim,
    float* __restrict__ carry)          // [B*C*D] x {pr,pi,yr,yi}
{
  const int c = blockIdx.x;
  const int b = blockIdx.y;
  const int d4 = threadIdx.x * 4;

  const size_t base = ((size_t)b * L + (size_t)c * T) * (size_t)D;
  const float* ga = Are + base;
  const float* gb = Aim + base;
  const float* gx = Xre + base;
  const float* gy = Xim + base;

  __shared__ float lds[2][4][S * D];  // double-buffered tiles, 32 KB

  async_stage(ga, gb, gx, gy,
              &lds[0][0][0], &lds[0][1][0], &lds[0][2][0], &lds[0][3][0], 0, d4);

  float yre[4], yim[4], pre[4], pim[4];
#pragma unroll
  for (int k = 0; k < 4; ++k) { yre[k] = 0.f; yim[k] = 0.f; pre[k] = 1.f; pim[k] = 0.f; }

  for (int s = 0; s < NST; ++s) {
    const int bb = s & 1;
    if (s + 1 < NST) {
      const int nb = bb ^ 1;
      async_stage(ga, gb, gx, gy,
                  &lds[nb][0][0], &lds[nb][1][0], &lds[nb][2][0], &lds[nb][3][0],
                  s + 1, d4);
      asm volatile("s_wait_asynccnt %0" :: "i"(4 * S) : "memory");
    } else {
      asm volatile("s_wait_asynccnt 0" ::: "memory");
    }
#pragma unroll
    for (int r = 0; r < S; ++r) {
      v4f ar4 = *(const v4f*)&lds[bb][0][r * D + d4];
      v4f ai4 = *(const v4f*)&lds[bb][1][r * D + d4];
      v4f xr4 = *(const v4f*)&lds[bb][2][r * D + d4];
      v4f xi4 = *(const v4f*)&lds[bb][3][r * D + d4];
#pragma unroll
      for (int k = 0; k < 4; ++k) {
        const float ar = ar4[k], ai = ai4[k], xr = xr4[k], xi = xi4[k];
        const float nr = fmaf(ar, yre[k], fmaf(-ai, yim[k], xr));
        const float ni = fmaf(ar, yim[k], fmaf(ai, yre[k], xi));
        yre[k] = nr; yim[k] = ni;
        const float qr = fmaf(ar, pre[k], -(ai * pim[k]));
        const float qi = fmaf(ar, pim[k], (ai * pre[k]));
        pre[k] = qr; pim[k] = qi;
      }
    }
  }

  v4f* cw = (v4f*)carry + ((size_t)(b * C + c) * D + d4);
#pragma unroll
  for (int k = 0; k < 4; ++k) {
    v4f v = {pre[k], pim[k], yre[k], yim[k]};
    cw[k] = v;
  }
}

// ---------------- Phase 2: cross-chunk scan of carries ---------------------
__global__ __launch_bounds__(256) void scan_carries(const float* __restrict__ carry,
                                                    float* __restrict__ prefix) {
  const int ch = blockIdx.x * 256 + threadIdx.x;  // 0..B*D-1
  const int b = ch / D;
  const int d = ch % D;
  float yr = 0.f, yi = 0.f;
  const v4f* cr = (const v4f*)carry;
  v2f* pw = (v2f*)prefix;
  for (int c = 0; c < C; ++c) {
    const size_t idx = (size_t)(b * C + c) * D + d;
    v2f p = {yr, yi};
    pw[idx] = p;                     // Y entering chunk c
    const v4f v = cr[idx];           // {prod_re, prod_im, ylast_re, ylast_im}
    const float nr = fmaf(v[0], yr, fmaf(-v[1], yi, v[2]));
    const float ni = fmaf(v[0], yi, fmaf(v[1], yr, v[3]));
    yr = nr; yi = ni;
  }
}

// ---------------- Phase 3: prefix-seeded re-scan, TDM staging --------------
// Each wave TDM-loads its 128-channel half of the tile and reads only that
// half, so s_wait_tensorcnt per wave suffices (no barriers).
__global__ __launch_bounds__(64) void scan_final(
    const float* __restrict__ Are, const float* __restrict__ Aim,
    const float* __restrict__ Xre, const float* __restrict__ Xim,
    const float* __restrict__ prefix,   // [B*C*D] x {yr,yi}
    float* __restrict__ out)            // [B,L,D,2]
{
  const int c = blockIdx.x;
  const int b = blockIdx.y;
  const int tid = threadIdx.x;
  const int d4 = tid * 4;
  const int w = tid >> 5;              // wave id 0/1 -> channel half
  const int col = (tid & 31) * 4;      // float offset within half-row

  const size_t base = ((size_t)b * L + (size_t)c * T) * (size_t)D;
  const float* gsrc[1];
  (void)gsrc;

  // [buf][array][wave-half][S*128] packed tiles, 32 KB total.
  __shared__ float lds[2][4][2][S * 128];

  const float* arrs0 = Are + base + (size_t)w * 128;
  const float* arrs1 = Aim + base + (size_t)w * 128;
  const float* arrs2 = Xre + base + (size_t)w * 128;
  const float* arrs3 = Xim + base + (size_t)w * 128;

  // Issue stage 0 (4 TDM loads per wave).
  tdm_load_tile(arrs0, (uint32_t)(uintptr_t)&lds[0][0][w][0]);
  tdm_load_tile(arrs1, (uint32_t)(uintptr_t)&lds[0][1][w][0]);
  tdm_load_tile(arrs2, (uint32_t)(uintptr_t)&lds[0][2][w][0]);
  tdm_load_tile(arrs3, (uint32_t)(uintptr_t)&lds[0][3][w][0]);

  float yre[4], yim[4];
  {
    const v2f* pr = (const v2f*)prefix + ((size_t)(b * C + c) * D + d4);
#pragma unroll
    for (int k = 0; k < 4; ++k) { v2f v = pr[k]; yre[k] = v[0]; yim[k] = v[1]; }
  }

  for (int s = 0; s < NST; ++s) {
    const int bb = s & 1;
    if (s + 1 < NST) {
      const int nb = bb ^ 1;
      const size_t soff = (size_t)(s + 1) * S * D;
      tdm_load_tile(arrs0 + soff, (uint32_t)(uintptr_t)&lds[nb][0][w][0]);
      tdm_load_tile(arrs1 + soff, (uint32_t)(uintptr_t)&lds[nb][1][w][0]);
      tdm_load_tile(arrs2 + soff, (uint32_t)(uintptr_t)&lds[nb][2][w][0]);
      tdm_load_tile(arrs3 + soff, (uint32_t)(uintptr_t)&lds[nb][3][w][0]);
      __builtin_amdgcn_s_wait_tensorcnt(4);  // TDM completes in order: stage s done
    } else {
      __builtin_amdgcn_s_wait_tensorcnt(0);
    }
#pragma unroll
    for (int r = 0; r < S; ++r) {
      const int t = s * S + r;
      v4f ar4 = *(const v4f*)&lds[bb][0][w][r * 128 + col];
      v4f ai4 = *(const v4f*)&lds[bb][1][w][r * 128 + col];
      v4f xr4 = *(const v4f*)&lds[bb][2][w][r * 128 + col];
      v4f xi4 = *(const v4f*)&lds[bb][3][w][r * 128 + col];
#pragma unroll
      for (int k = 0; k < 4; ++k) {
        const float ar = ar4[k], ai = ai4[k], xr = xr4[k], xi = xi4[k];
        const float nr = fmaf(ar, yre[k], fmaf(-ai, yim[k], xr));
        const float ni = fmaf(ar, yim[k], fmaf(ai, yre[k], xi));
        yre[k] = nr; yim[k] = ni;
      }
      v4f o0 = {yre[0], yim[0], yre[1], yim[1]};
      v4f o1 = {yre[2], yim[2], yre[3], yim[3]};
      const size_t tg = (size_t)c * T + (size_t)t;
      v4f* ow = (v4f*)out + ((((size_t)b * L + tg) * D + d4) >> 1);
      __builtin_nontemporal_store(o0, ow);      // keep A/X resident in L2
      __builtin_nontemporal_store(o1, ow + 1);
    }
  }
}

extern "C" void kernel_launch(void* const* d_in, const int* in_sizes, int n_in,
                              void* d_out, int out_size, void* d_ws, size_t ws_size,
                              hipStream_t stream) {
  (void)in_sizes; (void)n_in; (void)out_size; (void)ws_size;
  const float* Are = (const float*)d_in[0];
  const float* Aim = (const float*)d_in[1];
  const float* Xre = (const float*)d_in[2];
  const float* Xim = (const float*)d_in[3];
  float* out = (float*)d_out;

  float* carry = (float*)d_ws;                       // B*C*D*4 floats = 4 MiB
  float* prefix = carry + (size_t)B * C * D * 4;     // B*C*D*2 floats = 2 MiB

  dim3 grid(C, B), blk(64);
  scan_partial<<<grid, blk, 0, stream>>>(Are, Aim, Xre, Xim, carry);
  scan_carries<<<(B * D) / 256, 256, 0, stream>>>(carry, prefix);
  scan_final<<<grid, blk, 0, stream>>>(Are, Aim, Xre, Xim, prefix, out);
}